// Model1_32641751449662
// MI455X (gfx1250) — compile-verified
//
#include <hip/hip_runtime.h>

// Problem sizes (fixed by the reference)
#define NB   8      // batches
#define NN   8192   // points per batch
#define NS   4096   // FPS samples
#define NK   32     // kNN
#define NC16 16     // 64 channels as float4

typedef __attribute__((ext_vector_type(2))) float v2f;
typedef __attribute__((ext_vector_type(8))) float v8f;

// ---------------------------------------------------------------------------
// Kernel 1: farthest point sampling. One workgroup per batch, xyz + running
// distance array fully LDS-resident. 4096 sequential steps of
// (min-update, block argmax). Writes fps_idx (ws) and new_xyz (output 0).
// ---------------------------------------------------------------------------
__global__ __launch_bounds__(1024) void fps_kernel(
    const float* __restrict__ xyz, float* __restrict__ new_xyz,
    int* __restrict__ fps_idx) {
  extern __shared__ float smem[];
  float* xs   = smem;            // [NN]
  float* ys   = xs + NN;
  float* zs   = ys + NN;
  float* dist = zs + NN;
  float* redV = dist + NN;       // [32]
  int*   redI = (int*)(redV + 32);
  int*   curp = redI + 32;

  const int tid = threadIdx.x;
  const int b   = blockIdx.x;
  for (int i = tid; i < NN; i += 1024) {
    size_t o = ((size_t)b * NN + i) * 3;
    xs[i] = xyz[o + 0];
    ys[i] = xyz[o + 1];
    zs[i] = xyz[o + 2];
    dist[i] = 1.0e10f;
  }
  if (tid == 0) curp[0] = 0;
  __syncthreads();

  const int lane = tid & 31;
  const int wave = tid >> 5;
  const int ibase = tid * 8;

  for (int s = 0; s < NS; ++s) {
    int cur = curp[0];
    float cx = xs[cur], cy = ys[cur], cz = zs[cur];
    if (tid == 0) {
      size_t o = (size_t)b * NS + s;
      new_xyz[o * 3 + 0] = cx;
      new_xyz[o * 3 + 1] = cy;
      new_xyz[o * 3 + 2] = cz;
      fps_idx[o] = cur;
    }
    float bv = -1.0f;
    int bi = 0;
#pragma unroll
    for (int j = 0; j < 8; ++j) {
      int i = ibase + j;
      float dx = xs[i] - cx, dy = ys[i] - cy, dz = zs[i] - cz;
      float d = dx * dx + dy * dy + dz * dz;
      float nd = fminf(dist[i], d);
      dist[i] = nd;
      if (nd > bv) { bv = nd; bi = i; }   // strict > => first-occurrence argmax
    }
    // wave32 reduce; higher lanes hold strictly larger indices, so strict >
#pragma unroll
    for (int off = 16; off > 0; off >>= 1) {
      float ov = __shfl_down(bv, off);
      int   oi = __shfl_down(bi, off);
      if (ov > bv) { bv = ov; bi = oi; }
    }
    if (lane == 0) { redV[wave] = bv; redI[wave] = bi; }
    __syncthreads();
    if (tid < 32) {
      bv = redV[tid]; bi = redI[tid];
#pragma unroll
      for (int off = 16; off > 0; off >>= 1) {
        float ov = __shfl_down(bv, off);
        int   oi = __shfl_down(bi, off);
        if (ov > bv) { bv = ov; bi = oi; }
      }
      if (tid == 0) curp[0] = bi;
    }
    __syncthreads();
  }
}

// ---------------------------------------------------------------------------
// Kernel 2: fused distance + top-32 via V_WMMA_F32_16X16X4_F32.
// Block = 128 threads (4 waves), 64 queries/block, one batch per 64 blocks.
// Tile: rows M = 16 candidates, cols N = 16 queries, K = 4 (xyz + 0 pad).
//   A (16x4): lanes 0-15 supply {x,y}, lanes 16-31 {z,0} of rows c..c+15
//             (branchless: per-lane AoS base offset + ymask multiply).
//   B (4x16): column q = lane%16; lanes 0-15 {-2x,-2y}, hi lanes {-2z,0}.
//   C       : candidate squared norms in the D layout.
// => D[r] + |q|^2 = ||q||^2 + ||p||^2 - 2 q.p  (reference formula).
// xyz staged via gfx1250 async global->LDS DMA (ASYNCcnt), operands for the
// next chunk are preloaded so DS latency hides under the insert chain.
// ---------------------------------------------------------------------------
__device__ __forceinline__ void load_ops(const float* aos, const float* pn,
                                         int c, int aBase, float ymask, int hi8,
                                         v2f& a, v8f& cc) {
  a[0] = aos[3 * c + aBase];
  a[1] = aos[3 * c + aBase + 1] * ymask;   // hi lanes: garbage * 0 == 0 (finite)
  const float4* p4 = (const float4*)(pn + c + hi8);  // 32B-aligned
  float4 p40 = p4[0], p41 = p4[1];
  cc[0] = p40.x; cc[1] = p40.y; cc[2] = p40.z; cc[3] = p40.w;
  cc[4] = p41.x; cc[5] = p41.y; cc[6] = p41.z; cc[7] = p41.w;
}

__device__ __forceinline__ void insert8(v8f acc, float qn, int base,
                                        float (&ld)[NK], int (&li)[NK]) {
#pragma unroll
  for (int r = 0; r < 8; ++r) {
    float dv = acc[r] + qn;       // full squared distance
    int   iv = base + r;
    if (dv < ld[NK - 1]) {
      float cd = dv; int ci = iv;
#pragma unroll
      for (int j = 0; j < NK; ++j) {
        bool lt = cd < ld[j];
        float od = ld[j]; int oi = li[j];
        ld[j] = lt ? cd : od;
        li[j] = lt ? ci : oi;
        cd = lt ? od : cd;
        ci = lt ? oi : ci;
      }
    }
  }
}

__global__ __launch_bounds__(128) void knn_kernel(
    const float* __restrict__ xyz, const float* __restrict__ new_xyz,
    int* __restrict__ knn_idx, float* __restrict__ grouped_xyz) {
  extern __shared__ float smem[];
  float* aos = smem;               // [NN*3] xyz as-is (AoS)
  float* pn  = aos + NN * 3;       // [NN] candidate squared norms
  float* mD  = pn + NN;            // [4 waves][16 q][64]
  int*   mI  = (int*)(mD + 4 * 16 * 64);

  const int tid  = threadIdx.x;
  const int b    = blockIdx.x >> 6;          // 64 blocks per batch
  const int qblk = (blockIdx.x & 63) * 64;   // 64 queries per block

  // --- async DMA stage: 96KB AoS xyz -> LDS (6144 x b128, ASYNCcnt-tracked)
  const float4* gsrc = (const float4*)(xyz + (size_t)b * NN * 3);
  for (int i = tid; i < (NN * 3) / 4; i += 128) {
    unsigned lofs = (unsigned)(size_t)(const void*)(aos + i * 4);
    unsigned long long gaddr = (unsigned long long)(const void*)(gsrc + i);
    asm volatile("global_load_async_to_lds_b128 %0, %1, off"
                 :: "v"(lofs), "v"(gaddr) : "memory");
  }
  asm volatile("s_wait_asynccnt 0x0" ::: "memory");
  __syncthreads();

  for (int i = tid; i < NN; i += 128) {
    float x = aos[3 * i], y = aos[3 * i + 1], z = aos[3 * i + 2];
    pn[i] = x * x + y * y + z * z;
  }
  __syncthreads();

  const int wave = tid >> 5;
  const int lane = tid & 31;
  const int qm   = lane & 15;
  const int hi   = lane >> 4;                // 0: rows 0-7, 1: rows 8-15
  const int hi8  = hi * 8;
  const int q    = qblk + wave * 16 + qm;    // query within batch
  const size_t qg = (size_t)b * NS + q;

  const int   aBase = 3 * qm + 2 * hi;       // {x,y} for lo, {z,(x_next)*0} for hi
  const float ymask = hi ? 0.0f : 1.0f;

  float qx = new_xyz[qg * 3 + 0];
  float qy = new_xyz[qg * 3 + 1];
  float qz = new_xyz[qg * 3 + 2];
  const float qn = qx * qx + qy * qy + qz * qz;
  v2f bq;
  bq[0] = hi ? (-2.0f * qz) : (-2.0f * qx);
  bq[1] = hi ? 0.0f         : (-2.0f * qy);

  float ld[NK];
  int   li[NK];
#pragma unroll
  for (int j = 0; j < NK; ++j) { ld[j] = 3.0e38f; li[j] = 0; }

  // software-pipelined main loop: preload chunk c+16 before inserting chunk c
  v2f a; v8f cc;
  load_ops(aos, pn, 0, aBase, ymask, hi8, a, cc);
  int c = 0;
  for (; c < NN - 16; c += 16) {
    v8f acc = __builtin_amdgcn_wmma_f32_16x16x4_f32(
        /*neg_a=*/false, a, /*neg_b=*/false, bq,
        /*c_mod=*/(short)0, cc, /*reuse_a=*/false, /*reuse_b=*/false);
    v2f a2; v8f cc2;
    load_ops(aos, pn, c + 16, aBase, ymask, hi8, a2, cc2);
    insert8(acc, qn, c + hi8, ld, li);
    a = a2; cc = cc2;
  }
  {
    v8f acc = __builtin_amdgcn_wmma_f32_16x16x4_f32(
        false, a, false, bq, (short)0, cc, false, false);
    insert8(acc, qn, c + hi8, ld, li);
  }

  // dump both half-lists to LDS, then lanes 0-15 merge their query's pair
  float* wmDq = mD + (wave * 16 + qm) * 64 + hi * 32;
  int*   wmIq = mI + (wave * 16 + qm) * 64 + hi * 32;
#pragma unroll
  for (int j = 0; j < NK; ++j) { wmDq[j] = ld[j]; wmIq[j] = li[j]; }
  __syncthreads();

  if (lane < 16) {
    const float* dA = mD + (wave * 16 + qm) * 64;
    const int*   iA = mI + (wave * 16 + qm) * 64;
    const float* dB = dA + 32;
    const int*   iB = iA + 32;
    int i = 0, j = 0;
    const size_t outB = qg * NK;
    for (int k = 0; k < NK; ++k) {
      float da = dA[i], db = dB[j];
      int   ia = iA[i], ib = iB[j];
      bool takeA = (da < db) || (da == db && ia < ib);
      int sel = takeA ? ia : ib;
      if (takeA) ++i; else ++j;
      knn_idx[outB + k] = sel;
      grouped_xyz[(outB + k) * 3 + 0] = aos[3 * sel + 0];
      grouped_xyz[(outB + k) * 3 + 1] = aos[3 * sel + 1];
      grouped_xyz[(outB + k) * 3 + 2] = aos[3 * sel + 2];
    }
  }
}

// ---------------------------------------------------------------------------
// Gather kernels (float4 vectorized). Reads of `points` (2 MB/batch) are
// L2-resident; the grouped_points store (~268 MB) is the HBM floor.
// ---------------------------------------------------------------------------
__global__ __launch_bounds__(256) void gather_new_points(
    const float4* __restrict__ pts, const int* __restrict__ fps_idx,
    float4* __restrict__ out) {
  int t = blockIdx.x * 256 + threadIdx.x;      // NB*NS*16 = 524288
  int c = t & 15;
  int s = (t >> 4) & (NS - 1);
  int b = t >> 16;
  int n = fps_idx[b * NS + s];
  out[(size_t)(b * NS + s) * NC16 + c] = pts[((size_t)b * NN + n) * NC16 + c];
}

__global__ __launch_bounds__(256) void gather_grouped_points(
    const float4* __restrict__ pts, const int* __restrict__ knn_idx,
    float4* __restrict__ out) {
  size_t t = (size_t)blockIdx.x * 256 + threadIdx.x;  // NB*NS*NK*16 = 16777216
  int c = (int)(t & 15);
  size_t row = t >> 4;                                // (b*NS+s)*NK + k
  int n = knn_idx[row];
  size_t b = row >> 17;                               // NS*NK = 131072 rows/batch
  out[row * NC16 + c] = pts[(b * NN + n) * NC16 + c];
}

// ---------------------------------------------------------------------------
extern "C" void kernel_launch(void* const* d_in, const int* in_sizes, int n_in,
                              void* d_out, int out_size, void* d_ws, size_t ws_size,
                              hipStream_t stream) {
  (void)in_sizes; (void)n_in; (void)out_size; (void)ws_size;
  const float* xyz    = (const float*)d_in[0];   // [8,8192,3]
  const float* points = (const float*)d_in[1];   // [8,8192,64]
  float* out = (float*)d_out;

  // Output layout (flat, return order): new_xyz | new_points | grouped_xyz | grouped_points
  float* out_new_xyz        = out;                       // 98304
  float* out_new_points     = out + 98304;               // 2097152
  float* out_grouped_xyz    = out + 2195456;             // 3145728
  float* out_grouped_points = out + 5341184;             // 67108864

  int* fps_idx = (int*)d_ws;                             // [8,4096]
  int* knn_idx = fps_idx + NB * NS;                      // [8,4096,32]

  const int fps_smem = (4 * NN + 32) * (int)sizeof(float) + 33 * (int)sizeof(int);
  const int knn_smem = (NN * 3 + NN + 4 * 16 * 64) * (int)sizeof(float) +
                       (4 * 16 * 64) * (int)sizeof(int);
  (void)hipFuncSetAttribute((const void*)fps_kernel,
                            hipFuncAttributeMaxDynamicSharedMemorySize, fps_smem);
  (void)hipFuncSetAttribute((const void*)knn_kernel,
                            hipFuncAttributeMaxDynamicSharedMemorySize, knn_smem);

  fps_kernel<<<NB, 1024, fps_smem, stream>>>(xyz, out_new_xyz, fps_idx);
  knn_kernel<<<NB * (NS / 64), 128, knn_smem, stream>>>(
      xyz, out_new_xyz, knn_idx, out_grouped_xyz);
  gather_new_points<<<(NB * NS * NC16) / 256, 256, 0, stream>>>(
      (const float4*)points, fps_idx, (float4*)out_new_points);
  gather_grouped_points<<<(NB * NS * NK * NC16) / 256, 256, 0, stream>>>(
      (const float4*)points, knn_idx, (float4*)out_grouped_points);
}